// VQVAE_50242527429467
// MI455X (gfx1250) — compile-verified
//
#include <hip/hip_runtime.h>
#include <math.h>

#define B_     64
#define FIN    4
#define N_     2048
#define HID    512
#define ZD     128
#define KCODES 2048
#define ROWS   (B_ * N_)   /* 131072 tokens */

typedef float v2f __attribute__((ext_vector_type(2)));
typedef float v8f __attribute__((ext_vector_type(8)));

__device__ __forceinline__ float gelu_exact(float x) {
    // torch F.gelu default / jax approximate=False: 0.5*x*(1+erf(x/sqrt(2)))
    return 0.5f * x * (1.0f + erff(x * 0.70710678118654752f));
}

// ---------------------------------------------------------------------------
// Kernel 1: fused transpose + feature-norm + encoder layer 1 (K=4 -> VALU)
// out[r, j] = gelu( sum_f ((x[b,f,n]-mean[f])/std[f]) * w1[f,j] + b1[j] )
// ---------------------------------------------------------------------------
__global__ void enc1_kernel(const float* __restrict__ x,
                            const float* __restrict__ mean,
                            const float* __restrict__ stdv,
                            const float* __restrict__ w1,
                            const float* __restrict__ b1,
                            float* __restrict__ out) {
    int t = blockIdx.x * blockDim.x + threadIdx.x;   // over ROWS*HID
    int j = t & (HID - 1);
    int r = t >> 9;                                  // /HID
    int b = r >> 11;                                 // /N_
    int n = r & (N_ - 1);
    const float* xb = x + (size_t)b * FIN * N_ + n;
    float acc = b1[j];
#pragma unroll
    for (int f = 0; f < FIN; ++f) {
        float xn = (xb[(size_t)f * N_] - mean[f]) / stdv[f];
        acc += xn * w1[f * HID + j];
    }
    out[(size_t)r * HID + j] = gelu_exact(acc);
}

// ---------------------------------------------------------------------------
// WMMA fp32 GEMM:  C[M,N] = act(A[M,K] @ W[K,N] + bias)
// block = 256 threads = 8 waves, arranged (8/NWCOL) row-groups x NWCOL
// col-groups, NWCOL = N/64. Each wave owns a 16x64 tile; a block covers the
// FULL N width so every A row is streamed from HBM exactly once (redundant
// re-reads across waves of a block are served by L2/WGP$). Weights (<=1 MB)
// are L2-resident on the 192 MB L2.
// Grid = ROWS / (16 * 8/NWCOL).
// INDEXED: A row i is gathered as A[rowidx[i]] (codebook lookup for decoder).
// ---------------------------------------------------------------------------
template <int N, bool GELU, bool INDEXED>
__global__ void wmma_gemm_kernel(const float* __restrict__ A,
                                 const int*   __restrict__ rowidx,
                                 const float* __restrict__ W,
                                 const float* __restrict__ bias,
                                 float* __restrict__ C,
                                 int K) {
    constexpr int NWCOL = N / 64;          // waves across N (8 or 2)
    constexpr int NWROW = 8 / NWCOL;       // wave row-groups per block
    const int  wave = threadIdx.x >> 5;
    const int  lane = threadIdx.x & 31;
    const bool hi   = lane >= 16;
    const int  l15  = lane & 15;
    const int  m0   = blockIdx.x * (16 * NWROW) + (wave / NWCOL) * 16;
    const int  n0   = (wave % NWCOL) * 64;

    int ar = m0 + l15;
    if (INDEXED) ar = rowidx[ar];
    const float* arow = A + (size_t)ar * K + (hi ? 2 : 0);

    // B-fragment row pointers: lanes 0-15 carry K-offsets {0,1}, 16-31 {2,3}
    const float* wp0 = W + (size_t)(hi ? 2 : 0) * N + n0 + l15;
    const float* wp1 = W + (size_t)(hi ? 3 : 1) * N + n0 + l15;

    v8f acc0 = {}, acc1 = {}, acc2 = {}, acc3 = {};
    for (int k = 0; k < K; k += 4) {
        v2f a = *(const v2f*)(arow + k);
        const float* w0  = wp0 + (size_t)k * N;
        const float* w1p = wp1 + (size_t)k * N;
        v2f b0 = { w0[0],  w1p[0]  };
        v2f b1 = { w0[16], w1p[16] };
        v2f b2 = { w0[32], w1p[32] };
        v2f b3 = { w0[48], w1p[48] };
        acc0 = __builtin_amdgcn_wmma_f32_16x16x4_f32(false, a, false, b0, (short)0, acc0, false, false);
        acc1 = __builtin_amdgcn_wmma_f32_16x16x4_f32(false, a, false, b1, (short)0, acc1, false, false);
        acc2 = __builtin_amdgcn_wmma_f32_16x16x4_f32(false, a, false, b2, (short)0, acc2, false, false);
        acc3 = __builtin_amdgcn_wmma_f32_16x16x4_f32(false, a, false, b3, (short)0, acc3, false, false);
    }

    // C/D layout: vgpr v -> row m0 + v + (hi?8:0), col = tile + (lane&15)
    v8f accs[4] = { acc0, acc1, acc2, acc3 };
    const int row_base = m0 + (hi ? 8 : 0);
#pragma unroll
    for (int t = 0; t < 4; ++t) {
        int n = n0 + t * 16 + l15;
        float bv = bias[n];
#pragma unroll
        for (int v = 0; v < 8; ++v) {
            float val = accs[t][v] + bv;
            if (GELU) val = gelu_exact(val);
            C[(size_t)(row_base + v) * N + n] = val;
        }
    }
}

// ---------------------------------------------------------------------------
// Codebook squared norms: one wave per code
// ---------------------------------------------------------------------------
__global__ void code_norm_kernel(const float* __restrict__ codebook,
                                 float* __restrict__ c2) {
    int wave = threadIdx.x >> 5;
    int lane = threadIdx.x & 31;
    int c = blockIdx.x * 8 + wave;
    const float* row = codebook + (size_t)c * ZD;
    float s = 0.f;
    for (int j = lane; j < ZD; j += 32) s += row[j] * row[j];
#pragma unroll
    for (int off = 16; off > 0; off >>= 1) s += __shfl_xor(s, off, 32);
    if (lane == 0) c2[c] = s;
}

// ---------------------------------------------------------------------------
// Fused WMMA distance + argmin: wave holds 16 z-rows fully in registers,
// sweeps all 2048 codes in 16-wide tiles. d = c2[k] - 2*z.c (z2 constant
// per row, irrelevant for argmin). Cross-lane argmin via shfl_xor within
// each 16-lane half (rows 0-7 live in lanes 0-15, rows 8-15 in 16-31).
// ---------------------------------------------------------------------------
__global__ void vq_argmin_kernel(const float* __restrict__ z,
                                 const float* __restrict__ codebook,
                                 const float* __restrict__ c2,
                                 int* __restrict__ idx_out) {
    const int  wave = threadIdx.x >> 5;
    const int  lane = threadIdx.x & 31;
    const bool hi   = lane >= 16;
    const int  l15  = lane & 15;
    const int  m0   = blockIdx.x * 128 + wave * 16;

    const float* arow = z + (size_t)(m0 + l15) * ZD + (hi ? 2 : 0);
    v2f afrag[ZD / 4];
#pragma unroll
    for (int kk = 0; kk < ZD / 4; ++kk) afrag[kk] = *(const v2f*)(arow + kk * 4);

    float bestv[8];
    int   besti[8];
#pragma unroll
    for (int v = 0; v < 8; ++v) { bestv[v] = 3.4e38f; besti[v] = 0; }

    for (int c0 = 0; c0 < KCODES; c0 += 16) {
        v8f acc = {};
        const float* brow = codebook + (size_t)(c0 + l15) * ZD + (hi ? 2 : 0);
#pragma unroll
        for (int kk = 0; kk < ZD / 4; ++kk) {
            v2f bf = *(const v2f*)(brow + kk * 4);
            acc = __builtin_amdgcn_wmma_f32_16x16x4_f32(false, afrag[kk], false, bf, (short)0, acc, false, false);
        }
        int code = c0 + l15;
        float cn = c2[code];
#pragma unroll
        for (int v = 0; v < 8; ++v) {
            float d = cn - 2.0f * acc[v];
            if (d < bestv[v]) { bestv[v] = d; besti[v] = code; }
        }
    }
#pragma unroll
    for (int off = 1; off < 16; off <<= 1) {
#pragma unroll
        for (int v = 0; v < 8; ++v) {
            float ov = __shfl_xor(bestv[v], off, 32);
            int   oi = __shfl_xor(besti[v], off, 32);
            if (ov < bestv[v] || (ov == bestv[v] && oi < besti[v])) {
                bestv[v] = ov; besti[v] = oi;
            }
        }
    }
    if (l15 == 0) {
        int rb = m0 + (hi ? 8 : 0);
#pragma unroll
        for (int v = 0; v < 8; ++v) idx_out[rb + v] = besti[v];
    }
}

// ---------------------------------------------------------------------------
// Commitment loss, two-pass deterministic reduction (no float atomics)
// ---------------------------------------------------------------------------
__global__ void loss_partial_kernel(const float* __restrict__ z,
                                    const float* __restrict__ codebook,
                                    const int* __restrict__ idx,
                                    float* __restrict__ partials) {
    __shared__ float sdata[256];
    const size_t total = (size_t)ROWS * ZD;
    float s = 0.f;
    for (size_t i = (size_t)blockIdx.x * blockDim.x + threadIdx.x; i < total;
         i += (size_t)gridDim.x * blockDim.x) {
        int r = (int)(i >> 7);          // /ZD
        int j = (int)(i & (ZD - 1));
        float diff = z[i] - codebook[(size_t)idx[r] * ZD + j];
        s += diff * diff;
    }
    sdata[threadIdx.x] = s;
    __syncthreads();
    for (int off = 128; off > 0; off >>= 1) {
        if ((int)threadIdx.x < off) sdata[threadIdx.x] += sdata[threadIdx.x + off];
        __syncthreads();
    }
    if (threadIdx.x == 0) partials[blockIdx.x] = sdata[0];
}

__global__ void loss_final_kernel(const float* __restrict__ partials,
                                  float* __restrict__ out_loss) {
    __shared__ float sdata[256];
    float s = 0.f;
    for (int i = threadIdx.x; i < 1024; i += 256) s += partials[i];
    sdata[threadIdx.x] = s;
    __syncthreads();
    for (int off = 128; off > 0; off >>= 1) {
        if ((int)threadIdx.x < off) sdata[threadIdx.x] += sdata[threadIdx.x + off];
        __syncthreads();
    }
    if (threadIdx.x == 0)
        *out_loss = sdata[0] * (1.0f / ((float)ROWS * (float)ZD));
}

// ---------------------------------------------------------------------------
// Decoder output: wave-per-row 512->4 dot, denorm, transpose to [B,4,N]
// ---------------------------------------------------------------------------
__global__ void dec_out_kernel(const float* __restrict__ h,   // [ROWS, HID]
                               const float* __restrict__ w3,  // [HID, 4]
                               const float* __restrict__ b3,
                               const float* __restrict__ mean,
                               const float* __restrict__ stdv,
                               float* __restrict__ out) {      // [B, 4, N]
    int wave = threadIdx.x >> 5;
    int lane = threadIdx.x & 31;
    int r = blockIdx.x * 8 + wave;
    const float* hr = h + (size_t)r * HID;
    float a0 = 0.f, a1 = 0.f, a2 = 0.f, a3 = 0.f;
    const float4* wv = (const float4*)w3;
    for (int j = lane; j < HID; j += 32) {
        float hv = hr[j];
        float4 w = wv[j];
        a0 += hv * w.x; a1 += hv * w.y; a2 += hv * w.z; a3 += hv * w.w;
    }
#pragma unroll
    for (int off = 16; off > 0; off >>= 1) {
        a0 += __shfl_xor(a0, off, 32);
        a1 += __shfl_xor(a1, off, 32);
        a2 += __shfl_xor(a2, off, 32);
        a3 += __shfl_xor(a3, off, 32);
    }
    if (lane == 0) {
        int b = r >> 11, n = r & (N_ - 1);
        float* ob = out + (size_t)b * FIN * N_ + n;
        ob[0 * N_] = (a0 + b3[0]) * stdv[0] + mean[0];
        ob[1 * N_] = (a1 + b3[1]) * stdv[1] + mean[1];
        ob[2 * N_] = (a2 + b3[2]) * stdv[2] + mean[2];
        ob[3 * N_] = (a3 + b3[3]) * stdv[3] + mean[3];
    }
}

// ---------------------------------------------------------------------------
extern "C" void kernel_launch(void* const* d_in, const int* in_sizes, int n_in,
                              void* d_out, int out_size, void* d_ws, size_t ws_size,
                              hipStream_t stream) {
    const float* x        = (const float*)d_in[0];
    const float* mean     = (const float*)d_in[1];
    const float* stdv     = (const float*)d_in[2];
    const float* enc_w1   = (const float*)d_in[3];
    const float* enc_b1   = (const float*)d_in[4];
    const float* enc_w2   = (const float*)d_in[5];
    const float* enc_b2   = (const float*)d_in[6];
    const float* enc_w3   = (const float*)d_in[7];
    const float* enc_b3   = (const float*)d_in[8];
    const float* codebook = (const float*)d_in[9];
    const float* dec_w1   = (const float*)d_in[10];
    const float* dec_b1   = (const float*)d_in[11];
    const float* dec_w2   = (const float*)d_in[12];
    const float* dec_b2   = (const float*)d_in[13];
    const float* dec_w3   = (const float*)d_in[14];
    const float* dec_b3   = (const float*)d_in[15];

    char* ws = (char*)d_ws;
    const size_t BUF = (size_t)ROWS * HID * sizeof(float);  // 268 MB
    float* bufA = (float*)(ws);                              // h1 -> z -> d2
    float* bufB = (float*)(ws + BUF);                        // h2 -> d1
    float* c2   = (float*)(ws + 2 * BUF);
    int*   idx  = (int*)  (ws + 2 * BUF + KCODES * sizeof(float));
    float* part = (float*)(ws + 2 * BUF + KCODES * sizeof(float) + ROWS * sizeof(int));

    float* out      = (float*)d_out;
    float* out_loss = out + (size_t)B_ * FIN * N_;

    dim3 blk(256);

    // 1) h1 = gelu(xnorm @ enc_w1 + b1)           -> bufA
    enc1_kernel<<<(ROWS * HID) / 256, blk, 0, stream>>>(x, mean, stdv, enc_w1, enc_b1, bufA);
    // 2) h2 = gelu(h1 @ enc_w2 + b2)              -> bufB   (block tile 16x512)
    wmma_gemm_kernel<HID, true, false><<<ROWS / 16, blk, 0, stream>>>(
        bufA, nullptr, enc_w2, enc_b2, bufB, HID);
    // 3) z = h2 @ enc_w3 + b3                     -> bufA   (block tile 64x128)
    wmma_gemm_kernel<ZD, false, false><<<ROWS / 64, blk, 0, stream>>>(
        bufB, nullptr, enc_w3, enc_b3, bufA, HID);
    // 4) c2[k] = |codebook_k|^2
    code_norm_kernel<<<KCODES / 8, blk, 0, stream>>>(codebook, c2);
    // 5) idx = argmin_k |z - c_k|^2  (fused WMMA distances)
    vq_argmin_kernel<<<ROWS / 128, blk, 0, stream>>>(bufA, codebook, c2, idx);
    // 6) vq_loss = mean((z - codebook[idx])^2)
    loss_partial_kernel<<<1024, blk, 0, stream>>>(bufA, codebook, idx, part);
    loss_final_kernel<<<1, blk, 0, stream>>>(part, out_loss);
    // 7) d1 = gelu(codebook[idx] @ dec_w1 + b1)   -> bufB   (block tile 16x512)
    wmma_gemm_kernel<HID, true, true><<<ROWS / 16, blk, 0, stream>>>(
        codebook, idx, dec_w1, dec_b1, bufB, ZD);
    // 8) d2 = gelu(d1 @ dec_w2 + b2)              -> bufA   (block tile 16x512)
    wmma_gemm_kernel<HID, true, false><<<ROWS / 16, blk, 0, stream>>>(
        bufB, nullptr, dec_w2, dec_b2, bufA, HID);
    // 9) x_recon = (d2 @ dec_w3 + b3)*std + mean, transpose -> d_out
    dec_out_kernel<<<ROWS / 8, blk, 0, stream>>>(bufA, dec_w3, dec_b3, mean, stdv, out);
}